// TrimmedProcrustesLoss_61220463837478
// MI455X (gfx1250) — compile-verified
//
#include <hip/hip_runtime.h>

typedef __attribute__((ext_vector_type(2))) float v2f;
typedef __attribute__((ext_vector_type(8))) float v8f;

#define B_ 32
#define H_ 512
#define W_ 512
#define HW_ (H_ * W_)
#define N_ (B_ * HW_)
#define TRIM_ 0.2f
#define ALPHA_ 0.5f
#define EPS_ 1e-6f

// ---- monotone float <-> uint key (exact order-preserving) ----
__device__ __forceinline__ unsigned fkey(float f) {
    unsigned u = __float_as_uint(f);
    return (u & 0x80000000u) ? ~u : (u | 0x80000000u);
}
__device__ __forceinline__ float finv(unsigned k) {
    return (k & 0x80000000u) ? __uint_as_float(k & 0x7FFFFFFFu)
                             : __uint_as_float(~k);
}

// ---- wave32 sum via matrix engine ----
// A = ones(16x4), B holds one value per lane (64 slots, half zero).
// D = A*B has every row equal to the column sums of B, so VGPR0 of D holds
// each column sum twice across the two half-waves. Summing c[0] over all 32
// lanes gives 2 * (sum of all B entries) = 2 * (sum of lane values).
// Layout-agnostic: only needs A all-ones and B a bijective 64-slot store.
__device__ __forceinline__ float wave_sum(float v) {
    v2f a; a.x = 1.0f; a.y = 1.0f;
    v2f b; b.x = v;    b.y = 0.0f;
    v8f c = {};
    c = __builtin_amdgcn_wmma_f32_16x16x4_f32(false, a, false, b,
                                              (short)0, c, false, false);
    float s = c[0];
    s += __shfl_xor(s, 16, 32);
    s += __shfl_xor(s, 8, 32);
    s += __shfl_xor(s, 4, 32);
    s += __shfl_xor(s, 2, 32);
    s += __shfl_xor(s, 1, 32);
    return s * 0.5f;
}

// block-wide sum (blockDim.x multiple of 32, <= 256). scratch: >= 8 floats.
__device__ float block_sum(float v, float* scratch) {
    __syncthreads();                    // protect scratch reuse
    float w = wave_sum(v);
    if ((threadIdx.x & 31u) == 0u) scratch[threadIdx.x >> 5] = w;
    __syncthreads();
    if (threadIdx.x < 32u) {
        int nw = (int)(blockDim.x >> 5);
        float x = ((int)threadIdx.x < nw) ? scratch[threadIdx.x] : 0.0f;
        float t = wave_sum(x);
        if (threadIdx.x == 0u) scratch[0] = t;
    }
    __syncthreads();
    return scratch[0];
}

// ---- init workspace ----
__global__ void k_init(float* wsf) {
    wsf[blockIdx.x * blockDim.x + threadIdx.x] = 0.0f;
}

// ---- per-sample: exact lower median (radix select) + robust scale ----
// grid.x = 64: b = blockIdx.x>>1, tensor = blockIdx.x&1 (0=pred,1=tgt)
__global__ void k_sample_stats(const float* __restrict__ pred,
                               const float* __restrict__ tgt,
                               const float* __restrict__ mask,
                               float* __restrict__ stats /*[B][4]*/) {
    __shared__ unsigned hist[256];
    __shared__ float red[33];
    __shared__ unsigned sh_prefix, sh_rank;
    __shared__ float sh_m;

    int b = blockIdx.x >> 1;
    int t = blockIdx.x & 1;
    const float* xb = (t ? tgt : pred) + (size_t)b * HW_;
    const float* mb = mask + (size_t)b * HW_;
    int tid = (int)threadIdx.x;

    if (tid == 0) { sh_prefix = 0u; sh_rank = 0u; red[32] = 0.0f; }

    float ssum_loc = 0.0f;
    for (int r = 0; r < 4; ++r) {
        hist[tid] = 0u;
        __syncthreads();
        unsigned prefix = sh_prefix;
        int shift = 24 - 8 * r;
        for (int i = tid; i < HW_; i += 256) {
            float m = mb[i];
            if (r == 0) ssum_loc += m;
            if (m > 0.0f) {
                unsigned k = fkey(xb[i]);
                bool match = (r == 0) ||
                             ((k >> (shift + 8)) == (prefix >> (shift + 8)));
                if (match) atomicAdd(&hist[(k >> shift) & 255u], 1u);
            }
        }
        __syncthreads();
        if (tid == 0) {
            if (r == 0) {
                unsigned c = 0u;
                for (int j = 0; j < 256; ++j) c += hist[j];
                red[32] = (float)c;                         // stash cnt
                sh_rank = (c > 0u) ? ((c - 1u) >> 1) : 0u;  // lower median rank
            }
            unsigned rank = sh_rank, cum = 0u, bkt = 0u;
            for (int j = 0; j < 256; ++j) {
                unsigned cc = hist[j];
                if (cum + cc > rank) { bkt = (unsigned)j; rank -= cum; break; }
                cum += cc;
            }
            sh_rank = rank;
            sh_prefix = prefix | (bkt << shift);
        }
        __syncthreads();
    }

    float ssum = block_sum(ssum_loc, red);
    if (tid == 0) sh_m = (red[32] > 0.0f) ? finv(sh_prefix) : 0.0f;
    __syncthreads();
    float m = sh_m;

    float sq_loc = 0.0f;
    for (int i = tid; i < HW_; i += 256)
        sq_loc += mb[i] * fabsf(xb[i] - m);
    float sq = block_sum(sq_loc, red);

    if (tid == 0) {
        bool valid = ssum > 0.0f;
        float s = valid ? fmaxf(sq / fmaxf(ssum, EPS_), EPS_) : 1.0f;
        stats[b * 4 + 2 * t + 0] = m;
        stats[b * 4 + 2 * t + 1] = s;
    }
}

// ---- global residual histogram (one radix round) ----
__global__ void k_hist(const float* __restrict__ pred,
                       const float* __restrict__ tgt,
                       const float* __restrict__ mask,
                       const float* __restrict__ stats,
                       unsigned* __restrict__ hist_out,
                       const unsigned* __restrict__ sel,
                       int r, float* __restrict__ totalMask) {
    __shared__ unsigned hist[256];
    hist[threadIdx.x] = 0u;
    __syncthreads();
    unsigned prefix = (r > 0) ? sel[0] : 0u;
    int shift = 24 - 8 * r;
    float tm = 0.0f;
    int stride = (int)(gridDim.x * blockDim.x);
    for (int i = (int)(blockIdx.x * blockDim.x + threadIdx.x); i < N_; i += stride) {
        int b = i >> 18;  // HW_ = 2^18
        float mkv = mask[i];
        if (r == 0) tm += mkv;
        if (mkv > 0.0f) {
            float pn = (pred[i] - stats[b * 4 + 0]) / stats[b * 4 + 1];
            float tn = (tgt[i]  - stats[b * 4 + 2]) / stats[b * 4 + 3];
            unsigned k = fkey(fabsf(pn - tn));
            bool match = (r == 0) ||
                         ((k >> (shift + 8)) == (prefix >> (shift + 8)));
            if (match) atomicAdd(&hist[(k >> shift) & 255u], 1u);
        }
    }
    __syncthreads();
    if (hist[threadIdx.x]) atomicAdd(&hist_out[threadIdx.x], hist[threadIdx.x]);
    if (r == 0) {
        float w = wave_sum(tm);
        if ((threadIdx.x & 31u) == 0u) atomicAdd(totalMask, w);
    }
}

// ---- pick bucket for this radix round; round 0 also derives cnt/keep ----
// sel: [0]=prefix [1]=rank [2]=keep [3]=cnt [4]=zero_flag [5]=tkey
__global__ void k_scan(const unsigned* __restrict__ hist,
                       unsigned* __restrict__ sel, int r) {
    if (threadIdx.x == 0 && blockIdx.x == 0) {
        int shift = 24 - 8 * r;
        if (r == 0) {
            unsigned c = 0u;
            for (int j = 0; j < 256; ++j) c += hist[j];
            unsigned keep = (unsigned)floorf((float)c * (1.0f - TRIM_));
            sel[3] = c;
            sel[2] = keep;
            sel[4] = (c == 0u || keep == 0u) ? 1u : 0u;
            sel[1] = (keep > 0u) ? (keep - 1u) : 0u;
            sel[0] = 0u;
        }
        unsigned rank = sel[1], cum = 0u, bkt = 0u;
        for (int j = 0; j < 256; ++j) {
            unsigned cc = hist[j];
            if (cum + cc > rank) { bkt = (unsigned)j; rank -= cum; break; }
            cum += cc;
        }
        sel[1] = rank;
        sel[0] |= (bkt << shift);
        if (r == 3) sel[5] = sel[0];
    }
}

// ---- sum of residuals strictly below threshold + their count ----
__global__ void k_trim(const float* __restrict__ pred,
                       const float* __restrict__ tgt,
                       const float* __restrict__ mask,
                       const float* __restrict__ stats,
                       const unsigned* __restrict__ sel,
                       float* __restrict__ acc /*[0]=sum_lt [1]=cnt_lt*/) {
    unsigned tkey = sel[5];
    float s_lt = 0.0f, c_lt = 0.0f;
    int stride = (int)(gridDim.x * blockDim.x);
    for (int i = (int)(blockIdx.x * blockDim.x + threadIdx.x); i < N_; i += stride) {
        float mkv = mask[i];
        if (mkv > 0.0f) {
            int b = i >> 18;
            float pn = (pred[i] - stats[b * 4 + 0]) / stats[b * 4 + 1];
            float tn = (tgt[i]  - stats[b * 4 + 2]) / stats[b * 4 + 3];
            float rr = fabsf(pn - tn);
            if (fkey(rr) < tkey) { s_lt += rr; c_lt += 1.0f; }
        }
    }
    float ws_ = wave_sum(s_lt);
    float wc_ = wave_sum(c_lt);
    if ((threadIdx.x & 31u) == 0u) {
        atomicAdd(&acc[0], ws_);
        atomicAdd(&acc[1], wc_);
    }
}

// ---- gradient loss at one scale (exact subgrid, no tail threads) ----
__global__ void k_grad(const float* __restrict__ pred,
                       const float* __restrict__ tgt,
                       const float* __restrict__ mask,
                       const float* __restrict__ stats,
                       float* __restrict__ acc /*[0]=loss [1]=M*/,
                       int step, int S) {
    int e = (int)(blockIdx.x * blockDim.x + threadIdx.x);
    int per = S * S;
    int b = e / per;
    int rem = e - b * per;
    int y = rem / S;
    int xx = rem - y * S;
    size_t base = (size_t)b * HW_;
    size_t i00 = base + (size_t)(y * step) * W_ + (size_t)(xx * step);
    float mp = stats[b * 4 + 0], sp = stats[b * 4 + 1];
    float mt = stats[b * 4 + 2], st = stats[b * 4 + 3];
    float m0 = mask[i00];
    float d0 = m0 * ((pred[i00] - mp) / sp - (tgt[i00] - mt) / st);
    float lsum = 0.0f;
    if (xx + 1 < S) {
        size_t i01 = i00 + (size_t)step;
        float m1 = mask[i01];
        float d1 = m1 * ((pred[i01] - mp) / sp - (tgt[i01] - mt) / st);
        lsum += fabsf(d1 - d0) * (m1 * m0);
    }
    if (y + 1 < S) {
        size_t i10 = i00 + (size_t)step * W_;
        float m1 = mask[i10];
        float d1 = m1 * ((pred[i10] - mp) / sp - (tgt[i10] - mt) / st);
        lsum += fabsf(d1 - d0) * (m1 * m0);
    }
    float wl = wave_sum(lsum);
    float wm = wave_sum(m0);
    if ((threadIdx.x & 31u) == 0u) {
        atomicAdd(&acc[0], wl);
        atomicAdd(&acc[1], wm);
    }
}

// ---- final scalar combine ----
__global__ void k_final(const unsigned* __restrict__ sel,
                        const float* __restrict__ trimAcc,
                        const float* __restrict__ totalMask,
                        const float* __restrict__ gradAcc,
                        float* __restrict__ out) {
    if (threadIdx.x == 0 && blockIdx.x == 0) {
        float total = 0.0f;
        float tmask = totalMask[0];
        if (sel[4] == 0u && tmask != 0.0f) {
            float T = finv(sel[5]);
            float keepf = (float)sel[2];
            float trimmed = trimAcc[0] + (keepf - trimAcc[1]) * T;
            total = trimmed / tmask;
        }
        float reg = 0.0f;
        for (int s = 0; s < 4; ++s) {
            float M = gradAcc[2 * s + 1];
            if (M != 0.0f) reg += gradAcc[2 * s] / M;
        }
        out[0] = total + ALPHA_ * reg;
    }
}

extern "C" void kernel_launch(void* const* d_in, const int* in_sizes, int n_in,
                              void* d_out, int out_size, void* d_ws, size_t ws_size,
                              hipStream_t stream) {
    const float* pred = (const float*)d_in[0];
    const float* tgt  = (const float*)d_in[1];
    const float* mask = (const float*)d_in[2];
    float* wsf = (float*)d_ws;
    unsigned* wsu = (unsigned*)d_ws;

    float*    stats     = wsf;            // 128 floats
    float*    gradAcc   = wsf + 128;      // 8 floats
    float*    totalMask = wsf + 136;      // 1 float
    float*    trimAcc   = wsf + 137;      // 2 floats
    unsigned* hist      = wsu + 160;      // 4*256 uints
    unsigned* sel       = wsu + 1184;     // 6 uints
    float*    out       = (float*)d_out;

    hipLaunchKernelGGL(k_init, dim3(8), dim3(256), 0, stream, wsf);
    hipLaunchKernelGGL(k_sample_stats, dim3(64), dim3(256), 0, stream,
                       pred, tgt, mask, stats);
    for (int r = 0; r < 4; ++r) {
        hipLaunchKernelGGL(k_hist, dim3(512), dim3(256), 0, stream,
                           pred, tgt, mask, stats, hist + 256 * r, sel, r, totalMask);
        hipLaunchKernelGGL(k_scan, dim3(1), dim3(1), 0, stream,
                           hist + 256 * r, sel, r);
    }
    hipLaunchKernelGGL(k_trim, dim3(512), dim3(256), 0, stream,
                       pred, tgt, mask, stats, sel, trimAcc);
    for (int s = 0; s < 4; ++s) {
        int step = 1 << s;
        int S = 512 / step;
        int total = 32 * S * S;          // always a multiple of 256
        hipLaunchKernelGGL(k_grad, dim3(total / 256), dim3(256), 0, stream,
                           pred, tgt, mask, stats, gradAcc + 2 * s, step, S);
    }
    hipLaunchKernelGGL(k_final, dim3(1), dim3(1), 0, stream,
                       sel, trimAcc, totalMask, gradAcc, out);
}